// GPT2Attention_71725953843379
// MI455X (gfx1250) — compile-verified
//
#include <hip/hip_runtime.h>
#include <hip/hip_bf16.h>

typedef float v2f __attribute__((ext_vector_type(2)));
typedef float v4f __attribute__((ext_vector_type(4)));
typedef float v8f __attribute__((ext_vector_type(8)));

#define B_SZ   32
#define S_PAST 4095
#define S_TOT  4096
#define H_SZ   2048
#define NH_SZ  16
#define HD_SZ  128
#define H3     (3 * H_SZ)

// -------------------------------------------------------------------------
// Kernel 1/3: C[32,N] = A[32,2048] * W[2048,N] + bias[N]   (f32 WMMA)
// One wave per 16-col tile; M=32 -> two 16x16 accumulators sharing B.
// A 16x4 layout (ISA 7.12.2): lanes 0-15 rows 0-15 {K0,K1}, lanes 16-31 {K2,K3}.
// B 4x16: VGPR0 = rows K0|K2 striped over lanes, VGPR1 = rows K1|K3.
// C 16x16: VGPR r, lanes 0-15 -> (row r, col lane), lanes 16-31 -> (row r+8).
// -------------------------------------------------------------------------
__global__ __launch_bounds__(128)
void gemm32_wmma(const float* __restrict__ A,      // [32, 2048] row-major
                 const float* __restrict__ W,      // [2048, N] row-major
                 const float* __restrict__ bias,   // [N]
                 float* __restrict__ out,          // [32, N]
                 int N) {
    const int wave = threadIdx.x >> 5;
    const int lane = threadIdx.x & 31;
    const int hi   = lane >> 4;          // 0: K pair {0,1}, 1: K pair {2,3}
    const int lq   = lane & 15;          // row (A) / col (B)
    const int n0   = (blockIdx.x * 4 + wave) * 16;

    const float* Arow0 = A + (size_t)lq * H_SZ;          // rows 0..15
    const float* Arow1 = A + (size_t)(lq + 16) * H_SZ;   // rows 16..31

    v8f c0 = {};
    v8f c1 = {};
    for (int k0 = 0; k0 < H_SZ; k0 += 4) {
        const int kk = k0 + 2 * hi;
        v2f a0 = *(const v2f*)(Arow0 + kk);              // {K=kk, K=kk+1}
        v2f a1 = *(const v2f*)(Arow1 + kk);
        v2f bm;
        bm.x = W[(size_t)kk * N + n0 + lq];
        bm.y = W[(size_t)(kk + 1) * N + n0 + lq];
        c0 = __builtin_amdgcn_wmma_f32_16x16x4_f32(false, a0, false, bm,
                                                   (short)0, c0, false, false);
        c1 = __builtin_amdgcn_wmma_f32_16x16x4_f32(false, a1, false, bm,
                                                   (short)0, c1, false, false);
    }

    const float bv = bias[n0 + lq];
    const int row_off = hi ? 8 : 0;
#pragma unroll
    for (int r = 0; r < 8; ++r) {
        out[(size_t)(r + row_off) * N + n0 + lq]      = c0[r] + bv;   // rows 0..15
        out[(size_t)(r + row_off + 16) * N + n0 + lq] = c1[r] + bv;   // rows 16..31
    }
}

// -------------------------------------------------------------------------
// Kernel 2: decode attention, one workgroup per (b, h).
// 8 waves; lane owns head-dims [4*lane, 4*lane+3]; wave w owns keys s = w + 8i.
// Online (streaming) softmax in base-2; NT b128 loads of K/V (single use).
// -------------------------------------------------------------------------
__global__ __launch_bounds__(256)
void attn_decode(const float* __restrict__ past_key,   // [32, 4095, 2048]
                 const float* __restrict__ past_value, // [32, 4095, 2048]
                 const float* __restrict__ qkv,        // [32, 6144] (q|k_new|v_new)
                 float* __restrict__ ctx) {            // [32, 2048]
    const int h    = blockIdx.x;
    const int b    = blockIdx.y;
    const int tid  = threadIdx.x;
    const int wave = tid >> 5;
    const int lane = tid & 31;

    // (1/sqrt(128)) * log2(e): softmax done in base-2
    const float QS = 0.12751744f;

    const float* qbase = qkv + (size_t)b * H3 + h * HD_SZ + lane * 4;
    v4f q4 = *(const v4f*)qbase;
    q4 *= QS;

    const size_t kvoff = (size_t)b * S_PAST * H_SZ + (size_t)h * HD_SZ + lane * 4;
    const float* kp = past_key + kvoff;
    const float* vp = past_value + kvoff;
    const float* knew = qkv + (size_t)b * H3 + H_SZ + h * HD_SZ + lane * 4;
    const float* vnew = qkv + (size_t)b * H3 + 2 * H_SZ + h * HD_SZ + lane * 4;

    float m = -INFINITY, l = 0.0f;
    v4f acc = {};

#pragma unroll 4
    for (int s = wave; s < S_TOT; s += 8) {
        v4f k4, v4;
        if (s < S_PAST) {   // wave-uniform branch
            k4 = __builtin_nontemporal_load((const v4f*)(kp + (size_t)s * H_SZ));
            v4 = __builtin_nontemporal_load((const v4f*)(vp + (size_t)s * H_SZ));
        } else {
            k4 = *(const v4f*)knew;
            v4 = *(const v4f*)vnew;
        }
        float part = q4.x * k4.x + q4.y * k4.y + q4.z * k4.z + q4.w * k4.w;
#pragma unroll
        for (int off = 16; off > 0; off >>= 1)
            part += __shfl_xor(part, off, 32);
        // online softmax update (base-2)
        const float mn   = fmaxf(m, part);
        const float corr = __builtin_amdgcn_exp2f(m - mn);
        const float p    = __builtin_amdgcn_exp2f(part - mn);
        l = l * corr + p;
        acc.x = acc.x * corr + p * v4.x;
        acc.y = acc.y * corr + p * v4.y;
        acc.z = acc.z * corr + p * v4.z;
        acc.w = acc.w * corr + p * v4.w;
        m = mn;
    }

    // merge 8 per-wave partial softmaxes via LDS
    __shared__ float sm[8];
    __shared__ float sl[8];
    __shared__ __align__(16) float sacc[8][HD_SZ];
    *(v4f*)&sacc[wave][lane * 4] = acc;
    if (lane == 0) { sm[wave] = m; sl[wave] = l; }
    __syncthreads();

    if (tid < HD_SZ) {
        float M = sm[0];
#pragma unroll
        for (int w = 1; w < 8; ++w) M = fmaxf(M, sm[w]);
        float num = 0.0f, den = 0.0f;
#pragma unroll
        for (int w = 0; w < 8; ++w) {
            const float f = __builtin_amdgcn_exp2f(sm[w] - M);
            num += sacc[w][tid] * f;
            den += sl[w] * f;
        }
        ctx[(size_t)b * H_SZ + h * HD_SZ + tid] = num / den;
    }
}

// -------------------------------------------------------------------------
extern "C" void kernel_launch(void* const* d_in, const int* in_sizes, int n_in,
                              void* d_out, int out_size, void* d_ws, size_t ws_size,
                              hipStream_t stream) {
    const float* x          = (const float*)d_in[0];  // [32, 1, 2048]
    const float* past_key   = (const float*)d_in[1];  // [32, 4095, 2048]
    const float* past_value = (const float*)d_in[2];  // [32, 4095, 2048]
    const float* W_attn     = (const float*)d_in[3];  // [2048, 6144]
    const float* b_attn     = (const float*)d_in[4];  // [6144]
    const float* W_proj     = (const float*)d_in[5];  // [2048, 2048]
    const float* b_proj     = (const float*)d_in[6];  // [2048]
    float* out = (float*)d_out;                       // [32, 1, 2048]

    float* qkv_ws = (float*)d_ws;                     // [32, 6144]
    float* ctx_ws = qkv_ws + (size_t)B_SZ * H3;       // [32, 2048]

    // 1) qkv = x @ W_attn + b_attn   (N=6144 -> 96 blocks x 4 waves x 16 cols)
    gemm32_wmma<<<dim3(H3 / 64), dim3(128), 0, stream>>>(
        x, W_attn, b_attn, qkv_ws, H3);

    // 2) streaming softmax-attention over 4096 keys per (b, h)
    attn_decode<<<dim3(NH_SZ, B_SZ), dim3(256), 0, stream>>>(
        past_key, past_value, qkv_ws, ctx_ws);

    // 3) out = ctx @ W_proj + b_proj (N=2048 -> 32 blocks)
    gemm32_wmma<<<dim3(H_SZ / 64), dim3(128), 0, stream>>>(
        ctx_ws, W_proj, b_proj, out, H_SZ);
}